// IrrepsConvolution_46557445488729
// MI455X (gfx1250) — compile-verified
//
#include <hip/hip_runtime.h>

// ---------------------------------------------------------------------------
// Fused IrrepsConvolution for MI455X (gfx1250, wave32).
//   w = ssp(edge_emb @ W1/sqrt8) @ W2/8          <- GEMM2 via V_WMMA_F32_16X16X4_F32
//   message(e) built from x[src[e]], edge_attr[e], w[e]
//   out[dst[e]] += message(e) / denominator       <- global_atomic_add_f32 (L2-resident)
// One block = 256 threads = 8 waves; each wave owns 16 edges (block: 128 edges).
// ---------------------------------------------------------------------------

typedef float v2f __attribute__((ext_vector_type(2)));
typedef float v8f __attribute__((ext_vector_type(8)));

#define TPB        256
#define EDGES_PER_BLOCK 128
#define H_STRIDE   68     // floats; 16x68 h tile (8B aligned rows, conflict-friendly)
#define W_STRIDE   132    // floats; 16x132 w tile (half-wave bank offset of 32)
#define W1T_OFF    0      // 64 cols x 8 k, transposed+scaled: 2048 B
#define W2P_OFF    2048   // 32 k-pairs x 128 n of float2, scaled: 32768 B
#define WAVE_OFF   34816
#define WAVE_BYTES 8448   // max(16*68*4, 16*132*4) = 8448
#define SMEM_BYTES (WAVE_OFF + 8 * WAVE_BYTES)  // 102400 B

__device__ __forceinline__ float ssp_f(float v) {
    // (softplus(v) - ln2) * 1.679177, numerically stable
    const float LOG2F_  = 0.6931471805599453f;
    const float SSP_CST = 1.679177f;
    float sp = fmaxf(v, 0.0f) + log1pf(expf(-fabsf(v)));
    return (sp - LOG2F_) * SSP_CST;
}

__global__ __launch_bounds__(TPB) void irreps_conv_kernel(
    const float* __restrict__ x,          // (N,128)
    const float* __restrict__ edge_attr,  // (E,4)
    const float* __restrict__ edge_emb,   // (E,8)
    const int*   __restrict__ edge_index, // (2,E): [0]=dst, [1]=src
    const float* __restrict__ W1,         // (8,64)
    const float* __restrict__ W2,         // (64,128)
    const float* __restrict__ den,        // (1,)
    float*       __restrict__ out,        // (N,256), pre-zeroed
    int n_edges)
{
    extern __shared__ char smem_raw[];
    float* w1t = (float*)(smem_raw + W1T_OFF);  // w1t[n*8+k] = W1[k][n]/sqrt8
    v2f*   w2p = (v2f*)  (smem_raw + W2P_OFF);  // w2p[k2*128+n] = (W2[2k2][n],W2[2k2+1][n])/8

    const int tid  = threadIdx.x;
    const int lane = tid & 31;
    const int wave = tid >> 5;
    const int hi   = lane >> 4;   // half-wave select
    const int ln   = lane & 15;

    // ---- stage scaled weights into LDS --------------------------------------
    const float RSQRT8  = 0.35355339059327373f;
    const float RSQRT64 = 0.125f;
    for (int idx = tid; idx < 512; idx += TPB) {          // w1 transposed
        int k = idx & 7, n = idx >> 3;
        w1t[idx] = W1[k * 64 + n] * RSQRT8;
    }
    for (int idx = tid; idx < 4096; idx += TPB) {         // w2 as K-pairs
        int k2 = idx >> 7, n = idx & 127;
        v2f p;
        p.x = W2[(2 * k2)     * 128 + n] * RSQRT64;
        p.y = W2[(2 * k2 + 1) * 128 + n] * RSQRT64;
        w2p[idx] = p;
    }
    __syncthreads();

    float* hbuf = (float*)(smem_raw + WAVE_OFF + wave * WAVE_BYTES);
    const long long e0 = (long long)blockIdx.x * EDGES_PER_BLOCK + wave * 16;

    // ---- GEMM1 (K=8, scalar) + ssp activation; h tile -> LDS ----------------
    {
        long long eMy = e0 + ln;
        if (eMy >= n_edges) eMy = n_edges - 1;            // clamp (garbage h OK, masked later)
        const float4* er = (const float4*)(edge_emb + eMy * 8);
        float4 ea = er[0], eb = er[1];
        for (int i = 0; i < 32; ++i) {
            int n = hi * 32 + i;
            const float4* wc = (const float4*)(w1t + n * 8);   // broadcast read
            float4 wa = wc[0], wb = wc[1];
            float acc = ea.x * wa.x + ea.y * wa.y + ea.z * wa.z + ea.w * wa.w
                      + eb.x * wb.x + eb.y * wb.y + eb.z * wb.z + eb.w * wb.w;
            hbuf[ln * H_STRIDE + n] = ssp_f(acc);
        }
    }

    // ---- GEMM2: w(16x128) = h(16x64) @ W2s(64x128) via f32 WMMA -------------
    // A 16x4 f32 fragment: lane -> M = ln, VGPR0/1 -> K = 2*hi, 2*hi+1
    // B 4x16  f32 fragment: lane -> N = ln, VGPR0/1 -> K = 2*hi, 2*hi+1
    v8f wacc[8];
    const v8f vzero = {0.f, 0.f, 0.f, 0.f, 0.f, 0.f, 0.f, 0.f};
#pragma unroll
    for (int t = 0; t < 8; ++t) wacc[t] = vzero;

    for (int kc = 0; kc < 16; ++kc) {
        v2f a = *(const v2f*)(hbuf + ln * H_STRIDE + kc * 4 + 2 * hi);
        const v2f* brow = w2p + (kc * 2 + hi) * 128 + ln;
#pragma unroll
        for (int t = 0; t < 8; ++t) {
            v2f b = brow[t * 16];
            wacc[t] = __builtin_amdgcn_wmma_f32_16x16x4_f32(
                false, a, false, b, (short)0, wacc[t], false, false);
        }
    }

    // ---- spill w tile to per-wave LDS (reuse h buffer; DS is in-order) ------
    float* wbuf = hbuf;   // 16 x 132
#pragma unroll
    for (int t = 0; t < 8; ++t)
#pragma unroll
        for (int g = 0; g < 8; ++g)
            wbuf[(g + 8 * hi) * W_STRIDE + t * 16 + ln] = wacc[t][g];

    // ---- message construction + atomic scatter ------------------------------
    const int* e_dst = edge_index;            // row 0
    const int* e_src = edge_index + n_edges;  // row 1
    const float inv = 1.0f / den[0];
    const float RSQRT3 = 0.57735026918962576f;

    long long eAtt = e0 + ln;
    if (eAtt >= n_edges) eAtt = n_edges - 1;
    int    srcv = e_src[eAtt];
    int    dstv = e_dst[eAtt];
    float4 at   = *(const float4*)(edge_attr + eAtt * 4);

    const int u = lane;                       // lane <-> channel u in [0,32)
    for (int e = 0; e < 16; ++e) {
        if (e0 + e >= n_edges) break;         // uniform branch
        int   src = __shfl(srcv, e);
        int   dst = __shfl(dstv, e);
        float f0  = __shfl(at.x, e);
        float fv0 = __shfl(at.y, e);
        float fv1 = __shfl(at.z, e);
        float fv2 = __shfl(at.w, e);

        const float* xr = x + (long long)src * 128;
        float s  = xr[u];
        float v0 = xr[32 + 3 * u + 0];
        float v1 = xr[32 + 3 * u + 1];
        float v2 = xr[32 + 3 * u + 2];

        float w0  = wbuf[e * W_STRIDE +       u];
        float w1v = wbuf[e * W_STRIDE +  32 + u];
        float w2v = wbuf[e * W_STRIDE +  64 + u];
        float w3v = wbuf[e * W_STRIDE +  96 + u];

        float vdot = v0 * fv0 + v1 * fv1 + v2 * fv2;
        float* o = out + (long long)dst * 256;

        unsafeAtomicAdd(o + u,       inv * (w0 * s * f0));
        unsafeAtomicAdd(o + 32 + u,  inv * (w1v * vdot * RSQRT3));
        float c2 = inv * (w2v * s);
        unsafeAtomicAdd(o + 64 + 3 * u + 0, c2 * fv0);
        unsafeAtomicAdd(o + 64 + 3 * u + 1, c2 * fv1);
        unsafeAtomicAdd(o + 64 + 3 * u + 2, c2 * fv2);
        float c3 = inv * (w3v * f0);
        unsafeAtomicAdd(o + 160 + 3 * u + 0, c3 * v0);
        unsafeAtomicAdd(o + 160 + 3 * u + 1, c3 * v1);
        unsafeAtomicAdd(o + 160 + 3 * u + 2, c3 * v2);
    }
}

extern "C" void kernel_launch(void* const* d_in, const int* in_sizes, int n_in,
                              void* d_out, int out_size, void* d_ws, size_t ws_size,
                              hipStream_t stream) {
    const float* x         = (const float*)d_in[0];
    const float* edge_attr = (const float*)d_in[1];
    const float* edge_emb  = (const float*)d_in[2];
    const int*   edge_index= (const int*)  d_in[3];
    const float* W1        = (const float*)d_in[4];
    const float* W2        = (const float*)d_in[5];
    const float* den       = (const float*)d_in[6];
    float*       out       = (float*)d_out;

    const int E = in_sizes[2] / 8;   // edge_emb is (E,8)

    // output is accumulated atomically -> zero it every call (capturable memset node)
    hipMemsetAsync(d_out, 0, (size_t)out_size * sizeof(float), stream);

    // allow >64KB dynamic LDS (gfx1250 WGP has 320KB)
    hipFuncSetAttribute((const void*)irreps_conv_kernel,
                        hipFuncAttributeMaxDynamicSharedMemorySize, SMEM_BYTES);

    int grid = (E + EDGES_PER_BLOCK - 1) / EDGES_PER_BLOCK;
    irreps_conv_kernel<<<grid, TPB, SMEM_BYTES, stream>>>(
        x, edge_attr, edge_emb, edge_index, W1, W2, den, out, E);
}